// ALSTM_SIN_6622839570829
// MI455X (gfx1250) — compile-verified
//
#include <hip/hip_runtime.h>
#include <math.h>

typedef float v2f __attribute__((ext_vector_type(2)));
typedef float v8f __attribute__((ext_vector_type(8)));

#define T_TOTAL 2048
#define Bsz 64
#define Isz 128
#define Hsz 128
#define NCHUNK 8                   // T chunks (parallel via P/S scan decomposition)
#define CH_T (T_TOTAL / NCHUNK)    // 256 timesteps per chunk
#define SUB 16                     // timestep tile (WMMA M)
#define NSUB (CH_T / SUB)          // 16 sub-tiles per chunk

// Dynamic LDS layout (in floats)
#define OFF_W 0                          // 4 gates x 128 x 128 weights   (256 KB)
#define OFF_X0 (4 * Hsz * Isz)           // x tile double buffer 0        (8 KB)
#define OFF_X1 (OFF_X0 + SUB * Isz)      // x tile double buffer 1        (8 KB)
#define OFF_G (OFF_X1 + SUB * Isz)       // 4 gates x 16 x 128 activated  (32 KB)
#define OFF_B (OFF_G + 4 * SUB * Hsz)    // 4 x 128 biases                (2 KB)
#define SMEM_FLOATS (OFF_B + 4 * Hsz)    // 78336 floats = 313344 B (< 320 KB WGP LDS)

// ---------------------------------------------------------------------------
// Branch-free activations on the v_exp_f32 / v_rcp_f32 native path.
// tanh(x) = 1 - 2/(exp(2x)+1): saturates to +/-1 via exp overflow/underflow,
// propagates NaN, ~1e-6 accurate; no exec-mask divergence in the hot loop.
// ---------------------------------------------------------------------------
__device__ __forceinline__ float fast_tanh(float v) {
    return 1.0f - 2.0f * __builtin_amdgcn_rcpf(__expf(2.0f * v) + 1.0f);
}
__device__ __forceinline__ float fast_sigmoid(float v) {
    return __builtin_amdgcn_rcpf(1.0f + __expf(-v));
}

// ---------------------------------------------------------------------------
// Tensor Data Mover helpers (gfx1250). Guarded: falls back to manual staging
// if the builtin is absent (e.g. host pass).
// ---------------------------------------------------------------------------
#if defined(__has_builtin)
#if __has_builtin(__builtin_amdgcn_tensor_load_to_lds)
#define HAVE_TDM 1
#endif
#endif

#ifdef HAVE_TDM
typedef unsigned int v4u __attribute__((ext_vector_type(4)));
typedef int v4i __attribute__((ext_vector_type(4)));
typedef int v8i __attribute__((ext_vector_type(8)));

__device__ __forceinline__ void tdm_wait0() {
#if __has_builtin(__builtin_amdgcn_s_wait_tensorcnt)
    __builtin_amdgcn_s_wait_tensorcnt(0);
#else
    asm volatile("s_wait_tensorcnt 0x0" ::: "memory");
#endif
}
__device__ __forceinline__ void tdm_wait1() {
#if __has_builtin(__builtin_amdgcn_s_wait_tensorcnt)
    __builtin_amdgcn_s_wait_tensorcnt(1);
#else
    asm volatile("s_wait_tensorcnt 0x1" ::: "memory");
#endif
}

// Flat LDS pointer -> LDS byte address (aperture decode keeps addr[31:0]).
__device__ __forceinline__ unsigned lds_addr_of(const float* p) {
    return (unsigned)(uintptr_t)p;
}

// Async DMA of a 2-D tile (rows x cols 4-byte elements, global row stride in
// elements) into LDS, packed row-major. Tracked by TENSORcnt.
__device__ __forceinline__ void tdm_load_2d(unsigned lds_byte_addr,
                                            const void* gptr,
                                            unsigned rows, unsigned cols,
                                            unsigned row_stride_elems) {
    const unsigned long long ga = (unsigned long long)(uintptr_t)gptr;
    v4u g0;
    g0.x = 1u;                                       // count=1, user descriptor
    g0.y = lds_byte_addr;                            // lds_addr (bytes)
    g0.z = (unsigned)(ga & 0xffffffffull);           // global_addr[31:0]
    g0.w = (unsigned)((ga >> 32) & 0x01ffffffull)    // global_addr[56:32]
         | (2u << 30);                               // type = 2 ("image")
    v8i g1;
    g1[0] = (int)(2u << 16);                         // data_size=4B; no multicast/pad
    g1[1] = (int)((cols & 0xffffu) << 16);           // tensor_dim0[15:0]
    g1[2] = (int)(((cols >> 16) & 0xffffu)           // tensor_dim0[31:16]
         |  ((rows & 0xffffu) << 16));               // tensor_dim1[15:0]
    g1[3] = (int)(((rows >> 16) & 0xffffu)           // tensor_dim1[31:16]
         |  ((cols & 0xffffu) << 16));               // tile_dim0 = cols
    g1[4] = (int)(rows & 0xffffu);                   // tile_dim1 = rows, tile_dim2=0
    g1[5] = (int)row_stride_elems;                   // tensor_dim0_stride[31:0]
    g1[6] = 0;                                       // stride hi / dim1_stride
    g1[7] = 0;
    v4i g2 = {0, 0, 0, 0};                           // 2-D: groups 2/3 unused
    v4i g3 = {0, 0, 0, 0};
#if __clang_major__ >= 23
    v8i g4 = {0, 0, 0, 0, 0, 0, 0, 0};
    __builtin_amdgcn_tensor_load_to_lds(g0, g1, g2, g3, g4, 0);
#else
    __builtin_amdgcn_tensor_load_to_lds(g0, g1, g2, g3, 0);
#endif
}
#endif  // HAVE_TDM

// ---------------------------------------------------------------------------
// Kernel 1: per (chunk, batch) workgroup.
//   - TDM-stage this batch's 4 gate weight matrices in LDS (once)
//   - TDM double-buffer 16-timestep x tiles; per tile: f32 WMMA (16x16x4,
//     K-chain of 32) for 4 gates x 8 h-tiles (one per wave), activations,
//     local linear-scan reduction (P = prod f, S = running i*g contribution)
//   - emit (P, S) per (b, chunk, h); last chunk also emits o at t = T-1
// ---------------------------------------------------------------------------
__global__ void alstm_gates_scan(const float* __restrict__ x,
                                 const float* __restrict__ w_ig,
                                 const float* __restrict__ w_ii,
                                 const float* __restrict__ w_if,
                                 const float* __restrict__ w_io,
                                 const float* __restrict__ b_g,
                                 const float* __restrict__ b_i,
                                 const float* __restrict__ b_f,
                                 const float* __restrict__ b_o,
                                 float* __restrict__ wsP,
                                 float* __restrict__ wsS,
                                 float* __restrict__ wsO)
{
    extern __shared__ float smem[];
    const int chunk = blockIdx.x;      // 0..NCHUNK-1
    const int b     = blockIdx.y;      // 0..63
    const int tid   = threadIdx.x;     // 0..255 (8 wave32s)
    const int lane  = tid & 31;
    const int wave  = tid >> 5;        // 0..7 -> h-tile
    const int lo    = lane & 15;       // N / M index within half
    const int hi    = lane >> 4;       // half select (K pair / M+8)
    const int n0    = wave * 16;       // h-tile base
    const size_t wo = (size_t)b * Hsz * Isz;

#ifdef HAVE_TDM
    // ---- async stage: 4 weight matrices + first x tile (wave 0 issues) ----
    if (wave == 0) {
        tdm_load_2d(lds_addr_of(&smem[OFF_W + 0 * Hsz * Isz]), w_ig + wo, Hsz, Isz, Isz);
        tdm_load_2d(lds_addr_of(&smem[OFF_W + 1 * Hsz * Isz]), w_ii + wo, Hsz, Isz, Isz);
        tdm_load_2d(lds_addr_of(&smem[OFF_W + 2 * Hsz * Isz]), w_if + wo, Hsz, Isz, Isz);
        tdm_load_2d(lds_addr_of(&smem[OFF_W + 3 * Hsz * Isz]), w_io + wo, Hsz, Isz, Isz);
        tdm_load_2d(lds_addr_of(&smem[OFF_X0]),
                    x + (size_t)(chunk * CH_T) * (Bsz * Isz) + (size_t)b * Isz,
                    SUB, Isz, Bsz * Isz);
    }
    for (int idx = tid; idx < Hsz; idx += 256) {
        smem[OFF_B + 0 * Hsz + idx] = b_g[(size_t)b * Hsz + idx];
        smem[OFF_B + 1 * Hsz + idx] = b_i[(size_t)b * Hsz + idx];
        smem[OFF_B + 2 * Hsz + idx] = b_f[(size_t)b * Hsz + idx];
        smem[OFF_B + 3 * Hsz + idx] = b_o[(size_t)b * Hsz + idx];
    }
    // barrier happens at top of first loop iteration (after tensorcnt wait)
#else
    // ---- fallback: manual staging ----
    {
        const float4* src0 = (const float4*)(w_ig + wo);
        const float4* src1 = (const float4*)(w_ii + wo);
        const float4* src2 = (const float4*)(w_if + wo);
        const float4* src3 = (const float4*)(w_io + wo);
        float4* dst = (float4*)&smem[OFF_W];
        const int n4 = Hsz * Isz / 4;
        for (int idx = tid; idx < n4; idx += 256) {
            dst[0 * n4 + idx] = src0[idx];
            dst[1 * n4 + idx] = src1[idx];
            dst[2 * n4 + idx] = src2[idx];
            dst[3 * n4 + idx] = src3[idx];
        }
        for (int idx = tid; idx < Hsz; idx += 256) {
            smem[OFF_B + 0 * Hsz + idx] = b_g[(size_t)b * Hsz + idx];
            smem[OFF_B + 1 * Hsz + idx] = b_i[(size_t)b * Hsz + idx];
            smem[OFF_B + 2 * Hsz + idx] = b_f[(size_t)b * Hsz + idx];
            smem[OFF_B + 3 * Hsz + idx] = b_o[(size_t)b * Hsz + idx];
        }
    }
#endif

    float P = 1.0f, S = 0.0f, o_last = 0.0f;

    for (int sc = 0; sc < NSUB; ++sc) {
        const int t0 = chunk * CH_T + sc * SUB;

#ifdef HAVE_TDM
        // issue next tile into alternate buffer, then wait for current tile
        if (wave == 0) {
            if (sc + 1 < NSUB) {
                tdm_load_2d(lds_addr_of(&smem[OFF_X0 + ((sc + 1) & 1) * (SUB * Isz)]),
                            x + (size_t)(t0 + SUB) * (Bsz * Isz) + (size_t)b * Isz,
                            SUB, Isz, Bsz * Isz);
                tdm_wait1();   // current tile (and weights on sc==0) resident
            } else {
                tdm_wait0();
            }
        }
        __syncthreads();
        const float* xs = &smem[OFF_X0 + (sc & 1) * (SUB * Isz)];
#else
        for (int idx = tid; idx < SUB * Isz; idx += 256) {
            const int r = idx >> 7, c = idx & 127;
            smem[OFF_X0 + idx] = x[(size_t)(t0 + r) * (Bsz * Isz) + (size_t)b * Isz + c];
        }
        __syncthreads();
        const float* xs = &smem[OFF_X0];
#endif

        // ---- 16x16 output tile per wave, 4 gates, K = 128 in steps of 4 ----
        v8f acc0 = {}, acc1 = {}, acc2 = {}, acc3 = {};
        const float* xa = xs + lo * Isz + 2 * hi;                       // A: M=lo, K=2*hi+j
        const float* wb = &smem[OFF_W + (n0 + lo) * Isz + 2 * hi];      // B: N=lo, K=2*hi+j
        #pragma unroll 4
        for (int k = 0; k < Isz; k += 4) {
            v2f a  = *(const v2f*)(xa + k);
            v2f g0 = *(const v2f*)(wb + 0 * Hsz * Isz + k);
            v2f g1 = *(const v2f*)(wb + 1 * Hsz * Isz + k);
            v2f g2 = *(const v2f*)(wb + 2 * Hsz * Isz + k);
            v2f g3 = *(const v2f*)(wb + 3 * Hsz * Isz + k);
            acc0 = __builtin_amdgcn_wmma_f32_16x16x4_f32(false, a, false, g0, (short)0, acc0, false, false);
            acc1 = __builtin_amdgcn_wmma_f32_16x16x4_f32(false, a, false, g1, (short)0, acc1, false, false);
            acc2 = __builtin_amdgcn_wmma_f32_16x16x4_f32(false, a, false, g2, (short)0, acc2, false, false);
            acc3 = __builtin_amdgcn_wmma_f32_16x16x4_f32(false, a, false, g3, (short)0, acc3, false, false);
        }

        // ---- bias + activation, scatter D tile to LDS gact[gate][m][h] ----
        {
            const int hh = n0 + lo;
            const float bg = smem[OFF_B + 0 * Hsz + hh];
            const float bi = smem[OFF_B + 1 * Hsz + hh];
            const float bf = smem[OFF_B + 2 * Hsz + hh];
            const float bo = smem[OFF_B + 3 * Hsz + hh];
            #pragma unroll
            for (int r = 0; r < 8; ++r) {
                const int m = r + 8 * hi;  // D layout: vgpr r -> M = r + 8*hi, N = lane%16
                smem[OFF_G + 0 * SUB * Hsz + m * Hsz + hh] = fast_tanh(acc0[r] + bg);
                smem[OFF_G + 1 * SUB * Hsz + m * Hsz + hh] = fast_sigmoid(acc1[r] + bi);
                smem[OFF_G + 2 * SUB * Hsz + m * Hsz + hh] = fast_sigmoid(acc2[r] + bf);
                smem[OFF_G + 3 * SUB * Hsz + m * Hsz + hh] = fast_sigmoid(acc3[r] + bo);
            }
        }
        __syncthreads();

        // ---- local sequential scan over the 16 timesteps (thread = h) ----
        if (tid < Hsz) {
            const int hh = tid;
            #pragma unroll
            for (int m = 0; m < SUB; ++m) {
                const float gv = smem[OFF_G + 0 * SUB * Hsz + m * Hsz + hh];
                const float iv = smem[OFF_G + 1 * SUB * Hsz + m * Hsz + hh];
                const float fv = smem[OFF_G + 2 * SUB * Hsz + m * Hsz + hh];
                const float ov = smem[OFF_G + 3 * SUB * Hsz + m * Hsz + hh];
                S = fv * S + iv * gv;
                P *= fv;
                o_last = ov;
            }
        }
        __syncthreads();  // gact consumed; safe to reuse next iteration
    }

    if (tid < Hsz) {
        const size_t o = ((size_t)b * NCHUNK + chunk) * Hsz + tid;
        wsP[o] = P;
        wsS[o] = S;
        if (chunk == NCHUNK - 1) wsO[(size_t)b * Hsz + tid] = o_last;
    }
}

// ---------------------------------------------------------------------------
// Kernel 2: combine chunk partials sequentially per (b,h), h_fin = o*tanh(c),
// then the tiny FC head (128->32 tanh ->2) + log_softmax over 2 classes.
// ---------------------------------------------------------------------------
__global__ void alstm_head(const float* __restrict__ wsP,
                           const float* __restrict__ wsS,
                           const float* __restrict__ wsO,
                           const float* __restrict__ fc1_w,
                           const float* __restrict__ fc1_b,
                           const float* __restrict__ fc2_w,
                           const float* __restrict__ fc2_b,
                           float* __restrict__ out)
{
    __shared__ float hbuf[Hsz];
    __shared__ float a1[32];
    __shared__ float zz[2];
    const int b = blockIdx.x;
    const int tid = threadIdx.x;

    if (tid < Hsz) {
        float c = 0.0f;
        #pragma unroll
        for (int j = 0; j < NCHUNK; ++j) {
            const size_t o = ((size_t)b * NCHUNK + j) * Hsz + tid;
            c = wsP[o] * c + wsS[o];
        }
        hbuf[tid] = wsO[(size_t)b * Hsz + tid] * fast_tanh(c);
    }
    __syncthreads();

    if (tid < 32) {
        float acc = fc1_b[tid];
        #pragma unroll 8
        for (int h = 0; h < Hsz; ++h) acc += hbuf[h] * fc1_w[tid * Hsz + h];
        a1[tid] = fast_tanh(acc);
    }
    __syncthreads();

    if (tid < 2) {
        float z = fc2_b[tid];
        #pragma unroll
        for (int k = 0; k < 32; ++k) z += a1[k] * fc2_w[tid * 32 + k];
        zz[tid] = z;
    }
    __syncthreads();

    if (tid == 0) {
        const float m = fmaxf(zz[0], zz[1]);
        const float l = m + logf(__expf(zz[0] - m) + __expf(zz[1] - m));
        out[b * 2 + 0] = zz[0] - l;
        out[b * 2 + 1] = zz[1] - l;
    }
}

extern "C" void kernel_launch(void* const* d_in, const int* in_sizes, int n_in,
                              void* d_out, int out_size, void* d_ws, size_t ws_size,
                              hipStream_t stream) {
    (void)in_sizes; (void)n_in; (void)out_size; (void)ws_size;
    const float* x     = (const float*)d_in[0];
    const float* w_ig  = (const float*)d_in[1];
    const float* w_ii  = (const float*)d_in[2];
    const float* w_if  = (const float*)d_in[3];
    const float* w_io  = (const float*)d_in[4];
    // d_in[5..8] = w_hg/w_hi/w_hf/w_ho: multiplied by a zero hidden state in
    // the reference (its bug) -> identically zero contribution, skipped.
    const float* b_g   = (const float*)d_in[9];
    const float* b_i   = (const float*)d_in[10];
    const float* b_f   = (const float*)d_in[11];
    const float* b_o   = (const float*)d_in[12];
    const float* fc1_w = (const float*)d_in[13];
    const float* fc1_b = (const float*)d_in[14];
    const float* fc2_w = (const float*)d_in[15];
    const float* fc2_b = (const float*)d_in[16];

    float* ws  = (float*)d_ws;
    float* wsP = ws;                                    // [B][NCHUNK][H]
    float* wsS = ws + (size_t)Bsz * NCHUNK * Hsz;       // [B][NCHUNK][H]
    float* wsO = ws + (size_t)2 * Bsz * NCHUNK * Hsz;   // [B][H]

    const size_t smem_bytes = (size_t)SMEM_FLOATS * sizeof(float);
    hipFuncSetAttribute(reinterpret_cast<const void*>(alstm_gates_scan),
                        hipFuncAttributeMaxDynamicSharedMemorySize, (int)smem_bytes);

    alstm_gates_scan<<<dim3(NCHUNK, Bsz), 256, smem_bytes, stream>>>(
        x, w_ig, w_ii, w_if, w_io, b_g, b_i, b_f, b_o, wsP, wsS, wsO);

    alstm_head<<<Bsz, 128, 0, stream>>>(
        wsP, wsS, wsO, fc1_w, fc1_b, fc2_w, fc2_b, (float*)d_out);
}